// SSDBoxHead_37271726195288
// MI455X (gfx1250) — compile-verified
//
#include <hip/hip_runtime.h>
#include <hip/hip_bf16.h>
#include <stdint.h>

// ---------------- problem constants (match reference) ----------------
#define BATCH       32
#define NANCH       24564
#define NCLASS      81
#define NCLS        80          // foreground classes
#define NPAD        24576       // NANCH padded (float4-friendly)
#define APB         64          // anchors per block in kernel 1 (16 per wave32)
#define CENTER_VAR  0.1f
#define SIZE_VAR    0.2f
#define CONF_T      0.01f
#define IOU_T       0.45f
#define PRE_K       100
#define MAX_DET     100
#define EPSV        1e-8f

typedef float v2f __attribute__((ext_vector_type(2)));
typedef float v8f __attribute__((ext_vector_type(8)));

// ============================================================================
// Kernel 1: fused softmax (over 81 classes) + box decode.
//   Wave-cooperative: 16 anchors per wave32; lane pair (L, L+16) owns anchor.
//   Softmax denominators computed on the matrix pipe:
//   V_WMMA_F32_16X16X4_F32 with B = ones gives D[m][n] = sum_k A[m][k],
//   i.e. 16 row-sums per instruction, co-executing with the __expf VALU work.
// ============================================================================
__global__ __launch_bounds__(128)
void k_softmax_decode(const float* __restrict__ loc,
                      const float* __restrict__ conf,
                      const float* __restrict__ priors,
                      float* __restrict__ scores_t,
                      float* __restrict__ boxes) {
    __shared__ float tile[APB * NCLASS];   // 64*81*4 = 20.7 KB
    __shared__ float sums[4][16];          // per-wave row sums

    const int tid  = threadIdx.x;
    const int lane = tid & 31;
    const int w    = tid >> 5;             // wave id (0..3)
    const int b    = blockIdx.y;
    const int n0   = blockIdx.x * APB;
    const int cnt  = min(APB, NANCH - n0);

    // ---- coalesced tile load (cnt*81 floats; cnt is 64 or 52 -> /4 exact) ----
    const float* src = conf + ((size_t)b * NANCH + n0) * (size_t)NCLASS;
    const int total  = cnt * NCLASS;
    const int total4 = total >> 2;
    for (int i = tid; i < total4; i += 128)
        ((float4*)tile)[i] = ((const float4*)src)[i];
    for (int i = (total4 << 2) + tid; i < total; i += 128)
        tile[i] = src[i];
    __syncthreads();

    const int  aLoc   = (w << 4) + (lane & 15);       // anchor within tile
    const bool hiK    = (lane >= 16);
    const int  nValid = min(16, max(0, cnt - (w << 4)));
    const bool valid  = (lane & 15) < nValid;
    float* row = tile + aLoc * NCLASS;                // stride 81 -> conflict-free

    // ---- 1) max over 81 classes, split across the lane pair ----
    float m = -3.4e38f;
    if (valid) {
        const int k0 = hiK ? 40 : 0, k1 = hiK ? NCLASS : 40;
        for (int k = k0; k < k1; ++k) m = fmaxf(m, row[k]);
    }
    m = fmaxf(m, __shfl_xor(m, 16, 32));

    // ---- 2) exp in place (split across the lane pair) ----
    if (valid) {
        const int k0 = hiK ? 40 : 0, k1 = hiK ? NCLASS : 40;
        for (int k = k0; k < k1; ++k) row[k] = __expf(row[k] - m);
    }
    __syncthreads();

    // ---- 3) row sums on the matrix pipe: 21x v_wmma_f32_16x16x4_f32 ----
    // A layout (32-bit 16x4): lanes 0-15 hold (M=lane, K=0..1), lanes 16-31
    // hold (M=lane-16, K=2..3). B = ones => every D column equals the row sum.
    // Executed unconditionally by all lanes (EXEC must be all 1s for WMMA);
    // stale rows only affect their own D rows, whose results are never used.
    v8f acc = {0.f, 0.f, 0.f, 0.f, 0.f, 0.f, 0.f, 0.f};
    v2f bones; bones.x = 1.0f; bones.y = 1.0f;
    const int koff = hiK ? 2 : 0;
    #pragma unroll
    for (int k0 = 0; k0 < NCLASS; k0 += 4) {          // 84 >= 81, tail zeroed
        v2f a;
        const int ka = k0 + koff;
        a.x = (ka     < NCLASS) ? row[ka]     : 0.0f;
        a.y = (ka + 1 < NCLASS) ? row[ka + 1] : 0.0f;
        acc = __builtin_amdgcn_wmma_f32_16x16x4_f32(
                  false, a, false, bones, (short)0, acc, false, false);
    }
    // D layout: VGPR r, lanes 0-15 -> M=r; lanes 16-31 -> M=r+8 (all N equal)
    if (lane == 0) {
        #pragma unroll
        for (int r = 0; r < 8; ++r) sums[w][r] = acc[r];
    }
    if (lane == 16) {
        #pragma unroll
        for (int r = 0; r < 8; ++r) sums[w][8 + r] = acc[r];
    }
    __syncthreads();

    // ---- 4) normalize + transposed store + box decode ----
    if (valid) {
        const float inv = 1.0f / sums[w][lane & 15];
        const int   n   = n0 + aLoc;
        const size_t base = (size_t)b * NCLS * NPAD + n;
        const int c0 = hiK ? 41 : 1, c1 = hiK ? NCLASS : 41;
        for (int c = c0; c < c1; ++c)
            scores_t[base + (size_t)(c - 1) * NPAD] = row[c] * inv;

        if (!hiK) {   // one lane per anchor decodes the box
            const float4 l = ((const float4*)loc)[(size_t)b * NANCH + n];
            const float4 p = ((const float4*)priors)[n];
            const float cx = l.x * CENTER_VAR * p.z + p.x;
            const float cy = l.y * CENTER_VAR * p.w + p.y;
            const float bw = __expf(l.z * SIZE_VAR) * p.z;
            const float bh = __expf(l.w * SIZE_VAR) * p.w;
            float4 o; o.x = cx - 0.5f * bw; o.y = cy - 0.5f * bh;
                      o.z = cx + 0.5f * bw; o.w = cy + 0.5f * bh;
            ((float4*)boxes)[(size_t)b * NANCH + n] = o;
        }
    }
}

// ============================================================================
// Kernel 2: per-(b, class) top-100 selection.
//   - async-DMA the contiguous 96 KB score column into LDS
//     (gfx1250 GLOBAL_LOAD_ASYNC_TO_LDS_B128 + S_WAIT_ASYNCCNT, ASYNCcnt path)
//   - threshold at CONF_T, 100 stable-argmax extractions (tie -> lowest index,
//     matching lax.top_k), gather boxes for selected anchors.
// ============================================================================
__global__ __launch_bounds__(256)
void k_topk_per_class(const float* __restrict__ scores_t,
                      const float* __restrict__ boxes,
                      float* __restrict__ topk_scores,
                      float* __restrict__ topk_boxes) {
    extern __shared__ float sc[];              // NPAD floats (96 KB) dynamic LDS
    __shared__ float rv[256];
    __shared__ int   ri[256];

    const int tid = threadIdx.x;
    const int bc  = blockIdx.x;                // b * 80 + (class-1)
    const int b   = bc / NCLS;

    const float* src = scores_t + (size_t)bc * NPAD;

    // --- async global -> LDS bulk copy (per-lane B128, 24 uniform iterations)
    uint32_t lds0 = (uint32_t)(uintptr_t)(&sc[0]);   // low 32b of flat ptr == LDS offset
    for (int i = tid; i < (NPAD >> 2); i += 256) {
        uint32_t la = lds0 + ((uint32_t)i << 4);
        uint32_t go = (uint32_t)i << 4;
        asm volatile("global_load_async_to_lds_b128 %0, %1, %2"
                     :: "v"(la), "v"(go), "s"(src) : "memory");
    }
    asm volatile("s_wait_asynccnt 0" ::: "memory");
    __syncthreads();

    // --- threshold; pad region (>= NANCH) forced to 0
    for (int i = tid; i < NPAD; i += 256) {
        float v = (i < NANCH) ? sc[i] : 0.0f;
        sc[i] = (v > CONF_T) ? v : 0.0f;
    }
    __syncthreads();

    // --- 100 stable argmax extractions
    for (int it = 0; it < PRE_K; ++it) {
        float bv = -1.0f; int bi = NPAD;   // ascending scan keeps lowest tied index
        for (int i = tid; i < NPAD; i += 256) {
            float v = sc[i];
            if (v > bv) { bv = v; bi = i; }
        }
        rv[tid] = bv; ri[tid] = bi;
        __syncthreads();
        for (int s = 128; s > 0; s >>= 1) {
            if (tid < s) {
                float ov = rv[tid + s]; int oi = ri[tid + s];
                if (ov > rv[tid] || (ov == rv[tid] && oi < ri[tid])) { rv[tid] = ov; ri[tid] = oi; }
            }
            __syncthreads();
        }
        if (tid == 0) {
            const float val = rv[0];
            const int   idx = ri[0];
            topk_scores[(size_t)bc * PRE_K + it] = val;
            const int gi = (idx < NANCH) ? idx : 0;
            ((float4*)topk_boxes)[(size_t)bc * PRE_K + it] =
                ((const float4*)boxes)[(size_t)b * NANCH + gi];
            if (idx < NPAD) sc[idx] = -1.0f;
        }
        __syncthreads();
    }
}

// ============================================================================
// Kernel 3: greedy NMS over 100 boxes per (b, class). Sequential over i,
// parallel over j < i; IoU computed on the fly.
// ============================================================================
__global__ __launch_bounds__(128)
void k_nms(float* __restrict__ topk_scores,
           const float* __restrict__ topk_boxes) {
    const int tid = threadIdx.x;
    const int bc  = blockIdx.x;
    __shared__ float sx1[PRE_K], sy1[PRE_K], sx2[PRE_K], sy2[PRE_K];
    __shared__ float ss[PRE_K], sa[PRE_K];
    __shared__ int   keep[PRE_K];
    __shared__ int   flag;

    if (tid < PRE_K) {
        ss[tid] = topk_scores[(size_t)bc * PRE_K + tid];
        float4 bx = ((const float4*)topk_boxes)[(size_t)bc * PRE_K + tid];
        sx1[tid] = bx.x; sy1[tid] = bx.y; sx2[tid] = bx.z; sy2[tid] = bx.w;
        sa[tid] = fmaxf(bx.z - bx.x, 0.f) * fmaxf(bx.w - bx.y, 0.f);
    }
    __syncthreads();

    for (int i = 0; i < PRE_K; ++i) {
        if (tid == 0) flag = 0;
        __syncthreads();
        if (tid < i && keep[tid]) {
            float lx = fmaxf(sx1[tid], sx1[i]);
            float ly = fmaxf(sy1[tid], sy1[i]);
            float rx = fminf(sx2[tid], sx2[i]);
            float ry = fminf(sy2[tid], sy2[i]);
            float inter = fmaxf(rx - lx, 0.f) * fmaxf(ry - ly, 0.f);
            float iou = inter / (sa[tid] + sa[i] - inter + EPSV);
            if (iou > IOU_T) flag = 1;   // benign race: all writers store 1
        }
        __syncthreads();
        if (tid == 0) keep[i] = (ss[i] > 0.0f) && (flag == 0);
        __syncthreads();
    }
    if (tid < PRE_K)
        topk_scores[(size_t)bc * PRE_K + tid] = keep[tid] ? ss[tid] : 0.0f;
}

// ============================================================================
// Kernel 4: per-image top-100 over 80*100 = 8000 candidates; emits
// boxes [B,100,4], scores [B,100], labels [B,100] (as float).
// ============================================================================
__global__ __launch_bounds__(256)
void k_final_topk(const float* __restrict__ topk_scores,
                  const float* __restrict__ topk_boxes,
                  float* __restrict__ out_boxes,
                  float* __restrict__ out_scores,
                  float* __restrict__ out_labels) {
    const int tid = threadIdx.x;
    const int b   = blockIdx.x;
    __shared__ float sc4[NCLS * PRE_K];   // 8000 floats = 32 KB
    __shared__ float rv[256];
    __shared__ int   ri[256];

    for (int i = tid; i < NCLS * PRE_K; i += 256)
        sc4[i] = topk_scores[(size_t)b * NCLS * PRE_K + i];
    __syncthreads();

    for (int it = 0; it < MAX_DET; ++it) {
        float bv = -2.0f; int bi = 0;
        for (int i = tid; i < NCLS * PRE_K; i += 256) {
            float v = sc4[i];
            if (v > bv) { bv = v; bi = i; }
        }
        rv[tid] = bv; ri[tid] = bi;
        __syncthreads();
        for (int s = 128; s > 0; s >>= 1) {
            if (tid < s) {
                float ov = rv[tid + s]; int oi = ri[tid + s];
                if (ov > rv[tid] || (ov == rv[tid] && oi < ri[tid])) { rv[tid] = ov; ri[tid] = oi; }
            }
            __syncthreads();
        }
        if (tid == 0) {
            const int   f   = ri[0];
            const float val = rv[0];
            ((float4*)out_boxes)[b * MAX_DET + it] =
                ((const float4*)topk_boxes)[(size_t)b * NCLS * PRE_K + f];
            out_scores[b * MAX_DET + it] = val;
            out_labels[b * MAX_DET + it] = (float)(f / PRE_K + 1);
            sc4[f] = -3.0f;
        }
        __syncthreads();
    }
}

// ============================================================================
extern "C" void kernel_launch(void* const* d_in, const int* in_sizes, int n_in,
                              void* d_out, int out_size, void* d_ws, size_t ws_size,
                              hipStream_t stream) {
    (void)in_sizes; (void)n_in; (void)out_size; (void)ws_size;
    const float* loc    = (const float*)d_in[0];   // [B,N,4]
    const float* conf   = (const float*)d_in[1];   // [B,N,C]
    const float* priors = (const float*)d_in[2];   // [N,4]

    // workspace layout (floats): scores_t | boxes | topk_scores | topk_boxes
    float* scores_t = (float*)d_ws;                                    // B*80*NPAD
    float* boxes    = scores_t + (size_t)BATCH * NCLS * NPAD;          // B*N*4
    float* tks      = boxes    + (size_t)BATCH * NANCH * 4;            // B*80*100
    float* tkb      = tks      + (size_t)BATCH * NCLS * PRE_K;         // B*80*100*4

    float* out        = (float*)d_out;
    float* out_boxes  = out;                                           // B*100*4
    float* out_scores = out_boxes + (size_t)BATCH * MAX_DET * 4;       // B*100
    float* out_labels = out_scores + (size_t)BATCH * MAX_DET;          // B*100

    // Stage 1: softmax (WMMA row-sums) + decode
    dim3 g1((NANCH + APB - 1) / APB, BATCH);
    k_softmax_decode<<<g1, 128, 0, stream>>>(loc, conf, priors, scores_t, boxes);

    // Stage 2: per-class top-100 (96 KB dynamic LDS via async global->LDS DMA)
    const size_t smem2 = (size_t)NPAD * sizeof(float);
    (void)hipFuncSetAttribute((const void*)k_topk_per_class,
                              hipFuncAttributeMaxDynamicSharedMemorySize,
                              (int)smem2);
    k_topk_per_class<<<BATCH * NCLS, 256, smem2, stream>>>(scores_t, boxes, tks, tkb);

    // Stage 3: per-class NMS
    k_nms<<<BATCH * NCLS, 128, 0, stream>>>(tks, tkb);

    // Stage 4: per-image final top-100
    k_final_topk<<<BATCH, 256, 0, stream>>>(tks, tkb, out_boxes, out_scores, out_labels);
}